// SolverInTheLoop_17042430230548
// MI455X (gfx1250) — compile-verified
//
#include <hip/hip_runtime.h>
#include <stdint.h>

#define TPB 256
#define FORCE_BLOCKS 2048

namespace {
constexpr float SIGMA  = (float)(3.0 / 359.0 / 3.14159265358979323846);  // H = 1
constexpr float P_REF  = 100.0f;
constexpr float ETA0   = 0.01f;
constexpr float EPS    = 1e-8f;
constexpr float ETA_IJ = 2.0f * ETA0 * ETA0 / (ETA0 + ETA0 + EPS);       // eta_i == eta_j
}

__device__ __forceinline__ float pow5f(float t) { float t2 = t * t; return t2 * t2 * t; }
__device__ __forceinline__ float pow4f(float t) { float t2 = t * t; return t2 * t2; }

// Hardware f32 atomic-add (no-return form -> STOREcnt, no VGPR writeback).
__device__ __forceinline__ void atomic_add_f32(const float* base, unsigned byte_off, float val)
{
    asm volatile("global_atomic_add_f32 %0, %1, %2"
                 :: "v"(byte_off), "v"(val), "s"(base) : "memory");
}

// True LDS (address_space(3)) read of a staged 32-bit value.
__device__ __forceinline__ int lds_read_i32(unsigned lds_byte_addr)
{
    return *(volatile __attribute__((address_space(3))) int*)(unsigned)lds_byte_addr;
}

// ---------------------------------------------------------------------------
// Pass 1: rho[i] += W(|r_i - r_j|)  (one thread per edge, hw f32 atomic)
// ---------------------------------------------------------------------------
__global__ void sph_density(const float* __restrict__ r,
                            const int*   __restrict__ i_s,
                            const int*   __restrict__ j_s,
                            const float* __restrict__ rho,
                            int E)
{
    int e = blockIdx.x * TPB + threadIdx.x;
    if (e >= E) return;
    int i = i_s[e], j = j_s[e];
    float dx = r[3 * i + 0] - r[3 * j + 0];
    float dy = r[3 * i + 1] - r[3 * j + 1];
    float dz = r[3 * i + 2] - r[3 * j + 2];
    float d  = __builtin_amdgcn_sqrtf(dx * dx + dy * dy + dz * dz);  // v_sqrt_f32
    float q1 = fmaxf(0.0f, 1.0f - d);
    float q2 = fmaxf(0.0f, 2.0f - d);
    float q3 = fmaxf(0.0f, 3.0f - d);
    float w  = SIGMA * (pow5f(q3) - 6.0f * pow5f(q2) + 15.0f * pow5f(q1));
    atomic_add_f32(rho, (unsigned)i << 2, w);
}

// ---------------------------------------------------------------------------
// Pass 2: dudt[i] += c * (-p_ij * dr + eta_ij * (u_i - u_j))
//   (A·dr term is identically zero since v == u in the reference)
// Edge-index streams double-buffered into LDS via async-to-LDS (ASYNCcnt).
// ---------------------------------------------------------------------------
__global__ void sph_force(const float* __restrict__ r,
                          const float* __restrict__ v,
                          const float* __restrict__ rho,
                          const int*   __restrict__ i_s,
                          const int*   __restrict__ j_s,
                          const float* __restrict__ dudt,
                          int E, int niter)
{
    __shared__ int s_i[2 * TPB];
    __shared__ int s_j[2 * TPB];
    const int tid    = threadIdx.x;
    const int stride = gridDim.x * TPB;
    long long e      = (long long)blockIdx.x * TPB + tid;

    const unsigned li_base = (unsigned)(uintptr_t)&s_i[0];
    const unsigned lj_base = (unsigned)(uintptr_t)&s_j[0];

    // Prologue: stage tile 0 into buffer 0 (clamp tail lanes to a safe address).
    {
        long long en = (e < E) ? e : (long long)(E - 1);
        unsigned  voff = (unsigned)(en << 2);
        unsigned  li   = li_base + ((unsigned)tid << 2);
        unsigned  lj   = lj_base + ((unsigned)tid << 2);
        asm volatile("global_load_async_to_lds_b32 %0, %1, %2"
                     :: "v"(li), "v"(voff), "s"(i_s) : "memory");
        asm volatile("global_load_async_to_lds_b32 %0, %1, %2"
                     :: "v"(lj), "v"(voff), "s"(j_s) : "memory");
    }

    for (int k = 0; k < niter; ++k) {
        const unsigned buf  = (unsigned)((k & 1) * TPB + tid) << 2;
        const unsigned nbuf = (unsigned)(((k + 1) & 1) * TPB + tid) << 2;

        // Stream-prefetch two tiles ahead into L2.
        {
            long long ep = e + 2LL * stride;
            if (ep < E) {
                __builtin_prefetch(&i_s[ep], 0, 0);
                __builtin_prefetch(&j_s[ep], 0, 0);
            }
        }

        // Stage next tile into the other buffer (async, stays in flight).
        {
            long long en = e + stride;
            if (en >= E) en = E - 1;
            if (en < 0)  en = 0;
            unsigned voff = (unsigned)(en << 2);
            unsigned li   = li_base + nbuf;
            unsigned lj   = lj_base + nbuf;
            asm volatile("global_load_async_to_lds_b32 %0, %1, %2"
                         :: "v"(li), "v"(voff), "s"(i_s) : "memory");
            asm volatile("global_load_async_to_lds_b32 %0, %1, %2"
                         :: "v"(lj), "v"(voff), "s"(j_s) : "memory");
        }

        // Wait for the current tile's pair; the 2 just-issued stay in flight.
        asm volatile("s_wait_asynccnt 0x2" ::: "memory");

        int i = lds_read_i32(li_base + buf);   // ds_load_b32
        int j = lds_read_i32(lj_base + buf);   // ds_load_b32

        if (e < E) {
            float dx = r[3 * i + 0] - r[3 * j + 0];
            float dy = r[3 * i + 1] - r[3 * j + 1];
            float dz = r[3 * i + 2] - r[3 * j + 2];
            float d  = __builtin_amdgcn_sqrtf(dx * dx + dy * dy + dz * dz);

            float rhoi = rho[i], rhoj = rho[j];
            float pi   = P_REF * (rhoi - 1.0f);
            float pj   = P_REF * (rhoj - 1.0f);
            float pij  = (rhoj * pi + rhoi * pj) * __builtin_amdgcn_rcpf(rhoi + rhoj);
            float ri   = __builtin_amdgcn_rcpf(rhoi);
            float rj   = __builtin_amdgcn_rcpf(rhoj);
            float wvol = ri * ri + rj * rj;

            float q1 = fmaxf(0.0f, 1.0f - d);
            float q2 = fmaxf(0.0f, 2.0f - d);
            float q3 = fmaxf(0.0f, 3.0f - d);
            float gw = SIGMA * (-5.0f * pow4f(q3) + 30.0f * pow4f(q2) - 75.0f * pow4f(q1));
            float c  = wvol * gw * __builtin_amdgcn_rcpf(d + EPS);

            float ax = c * (ETA_IJ * (v[3 * i + 0] - v[3 * j + 0]) - pij * dx);
            float ay = c * (ETA_IJ * (v[3 * i + 1] - v[3 * j + 1]) - pij * dy);
            float az = c * (ETA_IJ * (v[3 * i + 2] - v[3 * j + 2]) - pij * dz);

            unsigned base_off = (unsigned)(3 * i) << 2;
            atomic_add_f32(dudt, base_off + 0u, ax);
            atomic_add_f32(dudt, base_off + 4u, ay);
            atomic_add_f32(dudt, base_off + 8u, az);
        }
        e += stride;
    }
}

// ---------------------------------------------------------------------------
// Pass 3: pack [rho, p, dudt(3), 0,0,0] per particle as two b128 stores.
// ---------------------------------------------------------------------------
__global__ void sph_pack(const float* __restrict__ rho,
                         const float* __restrict__ dudt,
                         float*       __restrict__ out,
                         int N)
{
    int n = blockIdx.x * TPB + threadIdx.x;
    if (n >= N) return;
    float rh = rho[n];
    float4 a = make_float4(rh, P_REF * (rh - 1.0f), dudt[3 * n + 0], dudt[3 * n + 1]);
    float4 b = make_float4(dudt[3 * n + 2], 0.0f, 0.0f, 0.0f);
    reinterpret_cast<float4*>(out)[2 * n + 0] = a;
    reinterpret_cast<float4*>(out)[2 * n + 1] = b;
}

extern "C" void kernel_launch(void* const* d_in, const int* in_sizes, int n_in,
                              void* d_out, int out_size, void* d_ws, size_t ws_size,
                              hipStream_t stream)
{
    (void)n_in; (void)out_size; (void)ws_size;
    const float* r   = (const float*)d_in[0];
    const float* v   = (const float*)d_in[1];
    const int*   i_s = (const int*)d_in[2];
    const int*   j_s = (const int*)d_in[3];
    const int N = in_sizes[0] / 3;
    const int E = in_sizes[2];

    float* rho  = (float*)d_ws;      // N floats
    float* dudt = rho + N;           // 3N floats

    hipMemsetAsync(d_ws, 0, (size_t)4 * (size_t)N * sizeof(float), stream);

    const int ebl = (E + TPB - 1) / TPB;
    sph_density<<<ebl, TPB, 0, stream>>>(r, i_s, j_s, rho, E);

    int fbl = FORCE_BLOCKS;
    if (fbl > ebl) fbl = ebl;
    const int stride = fbl * TPB;
    const int niter  = (E + stride - 1) / stride;
    sph_force<<<fbl, TPB, 0, stream>>>(r, v, rho, i_s, j_s, dudt, E, niter);

    const int nbl = (N + TPB - 1) / TPB;
    sph_pack<<<nbl, TPB, 0, stream>>>(rho, dudt, (float*)d_out, N);
}